// LinearConvolutionM2_18880676233728
// MI455X (gfx1250) — compile-verified
//
#include <hip/hip_runtime.h>
#include <hip/hip_bf16.h>

typedef float v2f __attribute__((ext_vector_type(2)));
typedef float v8f __attribute__((ext_vector_type(8)));

#define BB    8
#define CC    32
#define OR    8
#define HH    128
#define WW    128
#define KH    5
#define KW    5
#define K200  200            // OR * KH * KW
#define WPAD  (WW + 4)       // haloed row width (132)
#define ROWS  8              // output rows per workgroup
#define SROWS (ROWS + 4)     // staged (haloed) rows per s-plane (12)

// LDS offset of K-element kk=(s,ky,kx) within the staged x slab, for output
// row j=0 (tile j adds j*WPAD). Pure constexpr -> folds to literals under
// full unroll; the only runtime select is on lane bit 4 ("half").
__host__ __device__ constexpr int koff(int kk) {
    const int s  = kk / (KH * KW);
    const int ky = (kk - s * KH * KW) / KW;
    const int kx = kk - s * KH * KW - ky * KW;
    return (s * SROWS + ky) * WPAD + kx;
}

// ---------------------------------------------------------------------------
// Prep: per-channel GEMM weight matrix W_c[k=(s,ky,kx)][t(0..15)]
//   W_c[(s,ky,kx), t] = rotk[t, c, (s - t) mod 8, ky, kx]
// Columns t=8..15 zero-padded so WMMA N=16 is fully defined.
// ---------------------------------------------------------------------------
__global__ __launch_bounds__(256)
void se2_wmat_kernel(const float* __restrict__ kern, float* __restrict__ wmat) {
    const int c = blockIdx.x;
    for (int i = threadIdx.x; i < K200 * 16; i += 256) {
        const int k = i >> 4;
        const int t = i & 15;
        float val = 0.0f;
        if (t < OR) {
            const int s  = k / (KH * KW);
            const int r  = (k - s * KH * KW) / KW;
            const int q  = k - s * KH * KW - r * KW;
            const int tp = (s - t) & (OR - 1);
            const float ang = 6.283185307179586f * (float)t / (float)OR;
            const float ca = cosf(ang);
            const float sa = sinf(ang);
            const float yy = (float)(r - 2);
            const float xx = (float)(q - 2);
            const float ys = sa * xx + ca * yy;
            const float xs = ca * xx - sa * yy;
            const int iy = (int)rintf(ys + 2.0f);
            const int ix = (int)rintf(xs + 2.0f);
            if (iy >= 0 && iy < KH && ix >= 0 && ix < KW)
                val = kern[((c * OR + tp) * KH + iy) * KW + ix];
        }
        wmat[(c * K200 + k) * 16 + t] = val;
    }
}

// ---------------------------------------------------------------------------
// Main: one workgroup per (b, c, 8-row band). 8 waves; each wave owns a
// 16-pixel strip across 8 rows -> 8 accumulator tiles, 400 f32 WMMAs.
// B fragments (immediate-offset ds_load off per-lane base) are reused across
// all 8 row-tiles; A row-tiles are +528B immediate offsets off one
// cndmask-selected address pair per K-step.
// ---------------------------------------------------------------------------
__global__ __launch_bounds__(256)
void se2_conv_kernel(const float* __restrict__ x,
                     const float* __restrict__ wmat,
                     float* __restrict__ out) {
    __shared__ float xt[OR * SROWS * WPAD];  // 8 s-planes, 12 haloed rows, 132 cols
    __shared__ float wl[K200 * 16];          // per-channel B matrix (K x 16)

    const int bx = blockIdx.x;
    const int y0 = (bx & 15) << 3;           // band base row
    const int c  = (bx >> 4) & (CC - 1);
    const int b  = bx >> 9;

    const float* __restrict__ xc = x + (size_t)(b * CC + c) * OR * (HH * WW);

    // Stage haloed x slab (rows y0-2 .. y0+9), zero-filled out of range.
    for (int e = threadIdx.x; e < OR * SROWS * WPAD; e += 256) {
        const int s   = e / (SROWS * WPAD);
        const int rem = e - s * (SROWS * WPAD);
        const int rr  = rem / WPAD;
        const int col = rem - rr * WPAD;
        const int yy  = y0 + rr - 2;
        const int xx  = col - 2;
        float v = 0.0f;
        if (yy >= 0 && yy < HH && xx >= 0 && xx < WW)
            v = xc[(s * HH + yy) * WW + xx];
        xt[e] = v;
    }
    // Stage this channel's weight matrix (L2-resident source).
    const float* __restrict__ wg = wmat + (size_t)c * (K200 * 16);
    for (int i = threadIdx.x; i < K200 * 16; i += 256)
        wl[i] = wg[i];
    __syncthreads();

    const int wave = threadIdx.x >> 5;
    const int lane = threadIdx.x & 31;
    const int m    = lane & 15;              // A row (pixel) / B column (t)
    const int half = lane >> 4;              // K-pair selector (lane bit 4)
    const int x0   = wave << 4;              // 16-pixel strip base

    const float* __restrict__ xrow = xt + x0 + m;          // per-lane A base
    const float* __restrict__ wb   = wl + m + (half << 5); // per-lane B base

    v8f acc[ROWS];
#pragma unroll
    for (int j = 0; j < ROWS; ++j) acc[j] = (v8f){};

    // 50 K-steps x 8 row-tiles = 400 x V_WMMA_F32_16X16X4_F32 per wave.
#pragma unroll
    for (int k0 = 0; k0 < K200; k0 += 4) {
        const int oA = half ? koff(k0 + 2) : koff(k0);
        const int oB = half ? koff(k0 + 3) : koff(k0 + 1);
        const float* __restrict__ pa = xrow + oA;
        const float* __restrict__ pb = xrow + oB;

        v2f w;
        w.x = wb[(k0 << 4)];
        w.y = wb[(k0 << 4) + 16];

#pragma unroll
        for (int j = 0; j < ROWS; ++j) {
            v2f a;
            a.x = pa[j * WPAD];
            a.y = pb[j * WPAD];
            acc[j] = __builtin_amdgcn_wmma_f32_16x16x4_f32(
                false, a, false, w, (short)0, acc[j], false, false);
        }
    }

    // D layout: lane%16 = N = t ; VGPR r = row M = r + 8*half.
    // Each active lane writes 8 consecutive x positions -> b128 stores.
    const int t = m;
    if (t < OR) {
        float* __restrict__ po =
            out + ((((size_t)(b * CC + c) * OR + t) * HH + y0) * WW)
                + x0 + (half << 3);
#pragma unroll
        for (int j = 0; j < ROWS; ++j) {
#pragma unroll
            for (int r = 0; r < 8; ++r)
                po[j * WW + r] = acc[j][r];
        }
    }
}

// ---------------------------------------------------------------------------
extern "C" void kernel_launch(void* const* d_in, const int* in_sizes, int n_in,
                              void* d_out, int out_size, void* d_ws, size_t ws_size,
                              hipStream_t stream) {
    const float* x    = (const float*)d_in[0];   // [8,32,8,128,128] f32
    const float* kern = (const float*)d_in[1];   // [32,8,5,5] f32
    float* out  = (float*)d_out;                 // [8,32,8,128,128] f32
    float* wmat = (float*)d_ws;                  // 32*200*16 f32 = 400 KB scratch

    se2_wmat_kernel<<<CC, 256, 0, stream>>>(kern, wmat);
    se2_conv_kernel<<<BB * CC * (HH / ROWS), 256, 0, stream>>>(x, wmat, out);
}